// YoloLoss_8323646619845
// MI455X (gfx1250) — compile-verified
//
#include <hip/hip_runtime.h>
#include <math.h>

// ---------------------------------------------------------------------------
// YOLO loss for MI455X (gfx1250, wave32).
// Bandwidth-bound: ~250MB streamed -> ~11us ideal at 23.3 TB/s.
// One wave per anchor-cell (85 channels across lanes), GT boxes in LDS,
// softplus-form BCE (halves transcendentals), deterministic 2-stage reduce.
// Final reduction uses V_WMMA_F32_16X16X4_F32 (ones-matrix row-sum trick).
// ---------------------------------------------------------------------------

#define NCLS   80
#define CH     85
#define NBOX   70
#define NBATCH 16
#define WPB    8      // waves per 256-thread block (wave32)

typedef __attribute__((ext_vector_type(2))) float v2f;
typedef __attribute__((ext_vector_type(8))) float v8f;

__device__ __forceinline__ float softplusf(float x) {
  // log(1 + e^x), stable
  return fmaxf(x, 0.0f) + logf(1.0f + __expf(-fabsf(x)));
}
__device__ __forceinline__ float sigmoidf(float x) {
  return 1.0f / (1.0f + __expf(-x));
}

__global__ __launch_bounds__(256)
void yolo_scale_kernel(const float* __restrict__ conv,
                       const float* __restrict__ label,
                       const float* __restrict__ bboxes,   // [B,70,4]
                       float* __restrict__ partials,       // [gridDim.x*B][3]
                       int S, float stride,
                       float aw0, float ah0, float aw1, float ah1,
                       float aw2, float ah2)
{
  __shared__ float sbox[NBOX * 4];
  __shared__ float swsum[WPB][3];

  const int b = blockIdx.y;
  for (int i = threadIdx.x; i < NBOX * 4; i += blockDim.x)
    sbox[i] = bboxes[(size_t)b * (NBOX * 4) + i];
  __syncthreads();

  const int lane       = threadIdx.x & 31;
  const int wave       = threadIdx.x >> 5;
  const int cells      = S * S * 3;
  const int waveStride = gridDim.x * WPB;
  const float inputSize = stride * (float)S;
  const float invIn2    = 1.0f / (inputSize * inputSize);
  const float FOUR_OVER_PI2 = 0.40528473456935108577f;   // 4/pi^2

  const size_t batchBase = (size_t)b * (size_t)cells * CH;

  float accP = 0.0f;            // per-lane prob-loss partial
  float accCi = 0.0f;           // lane 0 only
  float accCf = 0.0f;           // lane 0 only

  for (int cell = blockIdx.x * WPB + wave; cell < cells; cell += waveStride) {
    const size_t base = batchBase + (size_t)cell * CH;

    // CDNA5 prefetch of next iteration's records (global_prefetch_b8)
    {
      const int nc = cell + waveStride;
      if (nc < cells) {
        const size_t nb = batchBase + (size_t)nc * CH;
        __builtin_prefetch(conv  + nb + lane, 0, 3);
        __builtin_prefetch(label + nb + lane, 0, 3);
      }
    }

    const int a  = cell % 3;
    const int xy = cell / 3;
    const int gx = xy % S;
    const int gy = xy / S;

    // Coalesced channel loads: lane handles ch = lane, lane+32, lane+64
    const int c1 = lane + 32, c2 = lane + 64;
    const float cv0 = conv [base + lane];
    const float lb0 = label[base + lane];
    const float cv1 = conv [base + c1];
    const float lb1 = label[base + c1];
    float cv2 = 0.0f, lb2 = 0.0f;
    if (c2 < CH) { cv2 = conv[base + c2]; lb2 = label[base + c2]; }

    // Broadcast box/conf channels to all lanes
    const float dx    = __shfl(cv0, 0), dy = __shfl(cv0, 1);
    const float dw    = __shfl(cv0, 2), dh = __shfl(cv0, 3);
    const float dconf = __shfl(cv0, 4);
    const float lx = __shfl(lb0, 0), ly = __shfl(lb0, 1);
    const float lw = __shfl(lb0, 2), lh = __shfl(lb0, 3);
    const float respond = __shfl(lb0, 4);

    // decode
    const float aw = (a == 0) ? aw0 : ((a == 1) ? aw1 : aw2);
    const float ah = (a == 0) ? ah0 : ((a == 1) ? ah1 : ah2);
    const float px = (sigmoidf(dx) + (float)gx) * stride;
    const float py = (sigmoidf(dy) + (float)gy) * stride;
    const float pw = __expf(dw) * aw;
    const float ph = __expf(dh) * ah;

    // ---- CIoU vs label box (identical on all lanes; cheap) ----
    const float b1x1 = px - pw * 0.5f, b1x2 = px + pw * 0.5f;
    const float b1y1 = py - ph * 0.5f, b1y2 = py + ph * 0.5f;
    const float b2x1 = lx - lw * 0.5f, b2x2 = lx + lw * 0.5f;
    const float b2y1 = ly - lh * 0.5f, b2y2 = ly + lh * 0.5f;
    const float n1x1 = fminf(b1x1, b1x2), n1x2 = fmaxf(b1x1, b1x2);
    const float n1y1 = fminf(b1y1, b1y2), n1y2 = fmaxf(b1y1, b1y2);
    const float n2x1 = fminf(b2x1, b2x2), n2x2 = fmaxf(b2x1, b2x2);
    const float n2y1 = fminf(b2y1, b2y2), n2y2 = fmaxf(b2y1, b2y2);
    const float area1 = (n1x2 - n1x1) * (n1y2 - n1y1);
    const float area2 = (n2x2 - n2x1) * (n2y2 - n2y1);
    const float iw = fmaxf(fminf(n1x2, n2x2) - fmaxf(n1x1, n2x1), 0.0f);
    const float ih = fmaxf(fminf(n1y2, n2y2) - fmaxf(n1y1, n2y1), 0.0f);
    const float interA = iw * ih;
    const float iou = interA / (area1 + area2 - interA + 1e-9f);
    const float ew = fmaxf(n1x2, n2x2) - fminf(n1x1, n2x1);
    const float eh = fmaxf(n1y2, n2y2) - fminf(n1y1, n2y1);
    const float cc = ew * ew + eh * eh;
    const float rho2 = (px - lx) * (px - lx) + (py - ly) * (py - ly);
    const float atd = atanf(pw / (ph + 1e-9f)) - atanf(lw / (lh + 1e-9f));
    const float v = FOUR_OVER_PI2 * atd * atd;
    const float alpha = v / (1.0f - iou + v);
    const float ciou = iou - rho2 / cc - alpha * v;
    const float ciouL = respond * (2.0f - lw * lh * invIn2) * (1.0f - ciou);

    // ---- max IoU vs 70 GT boxes, split across lanes ----
    float mx = 0.0f;
    const float parea = pw * ph;
    for (int j = lane; j < NBOX; j += 32) {
      const float tx = sbox[j * 4 + 0], ty = sbox[j * 4 + 1];
      const float tw = sbox[j * 4 + 2], th = sbox[j * 4 + 3];
      const float ix = fmaxf(fminf(b1x2, tx + tw * 0.5f) - fmaxf(b1x1, tx - tw * 0.5f), 0.0f);
      const float iy = fmaxf(fminf(b1y2, ty + th * 0.5f) - fmaxf(b1y1, ty - th * 0.5f), 0.0f);
      const float ia = ix * iy;
      mx = fmaxf(mx, ia / (parea + tw * th - ia));
    }
    #pragma unroll
    for (int off = 16; off; off >>= 1) mx = fmaxf(mx, __shfl_xor(mx, off));

    const float bgd = (1.0f - respond) * ((mx < 0.5f) ? 1.0f : 0.0f);
    // -log(sigmoid(x)) = softplus(-x); -log(1-sigmoid(x)) = softplus(x)
    const float confL = respond * softplusf(-dconf) + bgd * softplusf(dconf);

    // ---- class prob loss: softplus(x) - label*x per class ----
    float pl = 0.0f;
    if (lane >= 5) pl += softplusf(cv0) - lb0 * cv0;   // ch 5..31
    pl += softplusf(cv1) - lb1 * cv1;                  // ch 32..63
    if (c2 < CH) pl += softplusf(cv2) - lb2 * cv2;     // ch 64..84
    accP += respond * pl;
    if (lane == 0) { accCi += ciouL; accCf += confL; }
  }

  // wave reduce the per-lane prob accumulator
  #pragma unroll
  for (int off = 16; off; off >>= 1) accP += __shfl_xor(accP, off);
  if (lane == 0) {
    swsum[wave][0] = accCi;
    swsum[wave][1] = accCf;
    swsum[wave][2] = accP;
  }
  __syncthreads();
  if (threadIdx.x < 3) {
    float s = 0.0f;
    #pragma unroll
    for (int w = 0; w < WPB; ++w) s += swsum[w][threadIdx.x];
    partials[((size_t)blockIdx.y * gridDim.x + blockIdx.x) * 3 + threadIdx.x] = s;
  }
}

// Deterministic final reduction; the 256->1 stage uses f32 WMMA:
// D = A(16x4 data) x B(4x16 ones) + C  => rows of D are 4-way sums of A.
__global__ __launch_bounds__(256)
void yolo_reduce_kernel(const float* __restrict__ partials, int n,
                        float* __restrict__ out)
{
  __shared__ float sdata[256 * 3];
  float a0 = 0.0f, a1 = 0.0f, a2 = 0.0f;
  for (int i = threadIdx.x; i < n; i += 256) {
    a0 += partials[i * 3 + 0];
    a1 += partials[i * 3 + 1];
    a2 += partials[i * 3 + 2];
  }
  sdata[threadIdx.x * 3 + 0] = a0;
  sdata[threadIdx.x * 3 + 1] = a1;
  sdata[threadIdx.x * 3 + 2] = a2;
  __syncthreads();

  if (threadIdx.x < 32) {            // whole wave 0 active: EXEC all ones
    const int lane = threadIdx.x;
    v2f bones; bones.x = 1.0f; bones.y = 1.0f;
    float comp[3];
    #pragma unroll
    for (int c = 0; c < 3; ++c) {
      v8f acc = {};
      #pragma unroll
      for (int chunk = 0; chunk < 256; chunk += 64) {
        v2f amat;
        amat.x = sdata[(chunk + lane) * 3 + c];
        amat.y = sdata[(chunk + 32 + lane) * 3 + c];
        acc = __builtin_amdgcn_wmma_f32_16x16x4_f32(false, amat, false, bones,
                                                    (short)0, acc, false, false);
      }
      // sum D over M: 8 VGPRs per lane cover half the rows; pair with lane+16
      float s = acc[0] + acc[1] + acc[2] + acc[3] +
                acc[4] + acc[5] + acc[6] + acc[7];
      s += __shfl_xor(s, 16);
      comp[c] = s;
    }
    if (lane == 0) {
      const float invB = 1.0f / (float)NBATCH;   // mean over batch
      const float ci = comp[0] * invB;
      const float cf = comp[1] * invB;
      const float pr = comp[2] * invB;
      out[0] = ci + cf + pr;   // all_loss
      out[1] = ci;             // ciou_loss
      out[2] = cf;             // conf_loss
      out[3] = pr;             // prob_loss
    }
  }
}

extern "C" void kernel_launch(void* const* d_in, const int* in_sizes, int n_in,
                              void* d_out, int out_size, void* d_ws, size_t ws_size,
                              hipStream_t stream)
{
  // setup_inputs() order:
  const float* conv_l = (const float*)d_in[0];  // (16,19,19,255)
  const float* conv_m = (const float*)d_in[1];  // (16,38,38,255)
  const float* conv_s = (const float*)d_in[2];  // (16,76,76,255)
  const float* lab_s  = (const float*)d_in[3];  // (16,76,76,3,85)
  const float* lab_m  = (const float*)d_in[4];  // (16,38,38,3,85)
  const float* lab_l  = (const float*)d_in[5];  // (16,19,19,3,85)
  const float* tb     = (const float*)d_in[6];  // (16,70,4)
  float* out  = (float*)d_out;
  float* part = (float*)d_ws;                   // needs (136+34+9)*16*3 floats (~34KB)

  const int GX_S = 136, GX_M = 34, GX_L = 9;    // ~16 cells per wave each
  const int N_S = GX_S * NBATCH, N_M = GX_M * NBATCH, N_L = GX_L * NBATCH;
  float* pS = part;
  float* pM = part + (size_t)N_S * 3;
  float* pL = part + (size_t)(N_S + N_M) * 3;

  yolo_scale_kernel<<<dim3(GX_S, NBATCH), 256, 0, stream>>>(
      conv_s, lab_s, tb, pS, 76, 8.0f, 12.f, 16.f, 19.f, 36.f, 40.f, 28.f);
  yolo_scale_kernel<<<dim3(GX_M, NBATCH), 256, 0, stream>>>(
      conv_m, lab_m, tb, pM, 38, 16.0f, 36.f, 75.f, 76.f, 55.f, 72.f, 146.f);
  yolo_scale_kernel<<<dim3(GX_L, NBATCH), 256, 0, stream>>>(
      conv_l, lab_l, tb, pL, 19, 32.0f, 142.f, 110.f, 192.f, 243.f, 459.f, 401.f);

  yolo_reduce_kernel<<<1, 256, 0, stream>>>(part, N_S + N_M + N_L, out);
}